// TestNet_37108517437739
// MI455X (gfx1250) — compile-verified
//
#include <hip/hip_runtime.h>
#include <hip/hip_bf16.h>

#define GN 100000
#define GE 1200000
#define TOPK 512
#define EPS 1e-7f
#define BN_EPS 1e-5f
#define NEG_SLOPE 0.01f

typedef float v2f __attribute__((ext_vector_type(2)));
typedef float v8f __attribute__((ext_vector_type(8)));

// ---------------------------------------------------------------------------
// Layer-1 scalar kernels (C=16 is too small to be worth WMMA staging for K=3)
// ---------------------------------------------------------------------------

__global__ void k_lin1(const float* __restrict__ pos, const float* __restrict__ lw,
                       const float* __restrict__ lb, float* __restrict__ x1, int n) {
    int i = blockIdx.x * blockDim.x + threadIdx.x;
    if (i >= n) return;
    float p0 = pos[i * 3 + 0], p1 = pos[i * 3 + 1], p2 = pos[i * 3 + 2];
    #pragma unroll
    for (int c = 0; c < 16; ++c)
        x1[i * 16 + c] = p0 * lw[0 * 16 + c] + p1 * lw[1 * 16 + c] + p2 * lw[2 * 16 + c] + lb[c];
}

// msg = relu(x[src]) + EPS  (strictly positive -> int-bit atomicMax == float max)
template <int C>
__global__ void k_edge_max(const int* __restrict__ ei, const float* __restrict__ x,
                           int* __restrict__ segmax, int e) {
    int t = blockIdx.x * blockDim.x + threadIdx.x;
    if (t >= e) return;
    int s = ei[t], d = ei[GE + t];
    #pragma unroll 4
    for (int c = 0; c < C; ++c) {
        float m = fmaxf(x[s * C + c], 0.0f) + EPS;
        atomicMax(&segmax[d * C + c], __float_as_int(m));
    }
}

template <int C>
__global__ void k_edge_acc(const int* __restrict__ ei, const float* __restrict__ x,
                           const float* __restrict__ segmax, float* __restrict__ den,
                           float* __restrict__ num, int e) {
    int t = blockIdx.x * blockDim.x + threadIdx.x;
    if (t >= e) return;
    int s = ei[t], d = ei[GE + t];
    #pragma unroll 4
    for (int c = 0; c < C; ++c) {
        float m = fmaxf(x[s * C + c], 0.0f) + EPS;
        float a = __expf(m - segmax[d * C + c]);
        atomicAdd(&den[d * C + c], a);
        atomicAdd(&num[d * C + c], m * a);
    }
}

// softmax-normalize + residual + Linear(16->32) + BN stat accumulation
__global__ void k_node1(const float* __restrict__ x1, const float* __restrict__ num,
                        const float* __restrict__ den, const float* __restrict__ w1,
                        const float* __restrict__ b1, float* __restrict__ h1,
                        float* __restrict__ bnSum, float* __restrict__ bnSq, int n) {
    __shared__ float sS[32], sQ[32];
    if (threadIdx.x < 32) { sS[threadIdx.x] = 0.0f; sQ[threadIdx.x] = 0.0f; }
    __syncthreads();
    int i = blockIdx.x * blockDim.x + threadIdx.x;
    if (i < n) {
        float o[16];
        #pragma unroll
        for (int c = 0; c < 16; ++c)
            o[c] = num[i * 16 + c] / (den[i * 16 + c] + 1e-16f) + x1[i * 16 + c];
        #pragma unroll
        for (int j = 0; j < 32; ++j) {
            float s = b1[j];
            #pragma unroll
            for (int c = 0; c < 16; ++c) s += o[c] * w1[c * 32 + j];
            h1[i * 32 + j] = s;
            atomicAdd(&sS[j], s);
            atomicAdd(&sQ[j], s * s);
        }
    }
    __syncthreads();
    if (threadIdx.x < 32) {
        atomicAdd(&bnSum[threadIdx.x], sS[threadIdx.x]);
        atomicAdd(&bnSq[threadIdx.x], sQ[threadIdx.x]);
    }
}

template <int C>
__global__ void k_bnfin(const float* __restrict__ bnSum, const float* __restrict__ bnSq,
                        const float* __restrict__ g, const float* __restrict__ be,
                        float* __restrict__ bnA, float* __restrict__ bnB, int n) {
    int c = threadIdx.x;
    if (c >= C) return;
    float mu = bnSum[c] / (float)n;
    float var = bnSq[c] / (float)n - mu * mu;
    float inv = rsqrtf(var + BN_EPS);
    float a = g[c] * inv;
    bnA[c] = a;
    bnB[c] = be[c] - mu * a;
}

// BN+ReLU + Linear(32->16) + LeakyReLU
__global__ void k_node1b(const float* __restrict__ h1, const float* __restrict__ bnA,
                         const float* __restrict__ bnB, const float* __restrict__ w2,
                         const float* __restrict__ b2, float* __restrict__ x2, int n) {
    int i = blockIdx.x * blockDim.x + threadIdx.x;
    if (i >= n) return;
    float hb[32];
    #pragma unroll
    for (int j = 0; j < 32; ++j)
        hb[j] = fmaxf(h1[i * 32 + j] * bnA[j] + bnB[j], 0.0f);
    #pragma unroll
    for (int c = 0; c < 16; ++c) {
        float s = b2[c];
        #pragma unroll
        for (int j = 0; j < 32; ++j) s += hb[j] * w2[j * 16 + c];
        x2[i * 16 + c] = s > 0.0f ? s : NEG_SLOPE * s;
    }
}

// ---------------------------------------------------------------------------
// f32 WMMA GEMM: out[M,Nout] = op(A[M,Kd]) @ B[Kd,Nout] + bias, one wave/tile.
// Fragment layouts per CDNA5 ISA 7.12.2 (16x4 f32 A, 4x16 B, 16x16 f32 C/D).
// ---------------------------------------------------------------------------
template <bool ABN, bool LEAKY>
__global__ void k_wmma_gemm(const float* __restrict__ A, const float* __restrict__ B,
                            const float* __restrict__ bias, float* __restrict__ out,
                            int Kd, int Nout, const float* __restrict__ bnA,
                            const float* __restrict__ bnB, int numTiles) {
    int wave = blockIdx.x * (blockDim.x >> 5) + (threadIdx.x >> 5);
    if (wave >= numTiles) return;  // wave-uniform: EXEC stays all-ones for WMMA
    int lane = threadIdx.x & 31;
    int tilesN = Nout >> 4;
    int tm = wave / tilesN, tn = wave % tilesN;
    int l16 = lane & 15;
    int half = lane >> 4;
    int row = tm * 16 + l16;
    int col = tn * 16 + l16;
    float bval = bias[col];
    v8f acc = {bval, bval, bval, bval, bval, bval, bval, bval};
    int steps = Kd >> 2;
    for (int kk = 0; kk < steps; ++kk) {
        int k0 = kk * 4 + half * 2;
        v2f a = *(const v2f*)(A + (size_t)row * Kd + k0);
        if (ABN) {
            a.x = fmaxf(a.x * bnA[k0] + bnB[k0], 0.0f);
            a.y = fmaxf(a.y * bnA[k0 + 1] + bnB[k0 + 1], 0.0f);
        }
        v2f b;
        b.x = B[(size_t)k0 * Nout + col];
        b.y = B[(size_t)(k0 + 1) * Nout + col];
        acc = __builtin_amdgcn_wmma_f32_16x16x4_f32(false, a, false, b, (short)0, acc,
                                                    false, false);
    }
    #pragma unroll
    for (int v = 0; v < 8; ++v) {
        float r = acc[v];
        if (LEAKY) r = r > 0.0f ? r : NEG_SLOPE * r;
        out[(size_t)(tm * 16 + half * 8 + v) * Nout + col] = r;
    }
}

// out2 = num/(den+1e-16) + xl, written in place into `num`
__global__ void k_aggr2(float* __restrict__ num, const float* __restrict__ den,
                        const float* __restrict__ xl, int total) {
    int i = blockIdx.x * blockDim.x + threadIdx.x;
    if (i >= total) return;
    num[i] = num[i] / (den[i] + 1e-16f) + xl[i];
}

// BN stats for h2[N,128]: block covers 128 rows, 2 threads per channel
__global__ void k_bnstats128(const float* __restrict__ h, float* __restrict__ bnSum,
                             float* __restrict__ bnSq, int n) {
    __shared__ float shS[256], shQ[256];
    int c = threadIdx.x & 127;
    int rof = threadIdx.x >> 7;
    int r0 = blockIdx.x * 128;
    int rend = min(n, r0 + 128);
    float s = 0.0f, q = 0.0f;
    for (int r = r0 + rof; r < rend; r += 2) {
        float v = h[(size_t)r * 128 + c];
        s += v; q += v * v;
    }
    shS[threadIdx.x] = s; shQ[threadIdx.x] = q;
    __syncthreads();
    if (rof == 0) {
        s += shS[threadIdx.x + 128];
        q += shQ[threadIdx.x + 128];
        atomicAdd(&bnSum[c], s);
        atomicAdd(&bnSq[c], q);
    }
}

// ---------------------------------------------------------------------------
// TopK pooling
// ---------------------------------------------------------------------------
__global__ void k_wnorm(const float* __restrict__ w, float* __restrict__ invnorm) {
    __shared__ float red[64];
    float v = w[threadIdx.x];
    red[threadIdx.x] = v * v;
    __syncthreads();
    for (int s = 32; s > 0; s >>= 1) {
        if (threadIdx.x < s) red[threadIdx.x] += red[threadIdx.x + s];
        __syncthreads();
    }
    if (threadIdx.x == 0) invnorm[0] = rsqrtf(red[0]);
}

__global__ void k_score(const float* __restrict__ x3, const float* __restrict__ w,
                        const float* __restrict__ invnorm, float* __restrict__ scores,
                        unsigned* __restrict__ keys, int n) {
    int i = blockIdx.x * blockDim.x + threadIdx.x;
    if (i >= n) return;
    float s = 0.0f;
    #pragma unroll 8
    for (int c = 0; c < 64; ++c) s += x3[(size_t)i * 64 + c] * w[c];
    s = tanhf(s * invnorm[0]);
    scores[i] = s;
    unsigned u = __float_as_uint(s);
    keys[i] = (u & 0x80000000u) ? ~u : (u | 0x80000000u);
}

// single block: 32-pass radix select -> gather -> bitonic sort -> gated output
__global__ void __launch_bounds__(1024)
k_topk(const unsigned* __restrict__ keys, const float* __restrict__ scores,
       const float* __restrict__ x3, float* __restrict__ out, int n) {
    __shared__ unsigned red[1024];
    __shared__ unsigned skey[1024];
    __shared__ int sidx[1024];
    __shared__ unsigned sP;
    __shared__ int sK, sCur, sTot;
    int tid = threadIdx.x;
    if (tid == 0) { sP = 0u; sK = TOPK; sCur = 0; }
    __syncthreads();
    // radix select: find 512th largest key
    for (int b = 31; b >= 0; --b) {
        unsigned prefMask = ~((1u << b) - 1u);
        unsigned Pc = sP | (1u << b);
        unsigned cnt = 0;
        for (int i = tid; i < n; i += 1024)
            if ((keys[i] & prefMask) == Pc) cnt++;
        red[tid] = cnt;
        __syncthreads();
        for (int s = 512; s > 0; s >>= 1) {
            if (tid < s) red[tid] += red[tid + s];
            __syncthreads();
        }
        if (tid == 0) {
            if ((int)red[0] >= sK) sP = Pc;
            else sK -= (int)red[0];
        }
        __syncthreads();
    }
    unsigned kth = sP;
    // gather candidates (>= kth). Guaranteed >= 512 of them.
    for (int i = tid; i < n; i += 1024) {
        unsigned k = keys[i];
        if (k >= kth) {
            int p = atomicAdd(&sCur, 1);
            if (p < 1024) { skey[p] = k; sidx[p] = i; }
        }
    }
    __syncthreads();
    if (tid == 0) sTot = min(sCur, 1024);
    __syncthreads();
    for (int i = tid; i < 1024; i += 1024) {
        if (i >= sTot) { skey[i] = 0u; sidx[i] = 0x7fffffff; }
    }
    __syncthreads();
    // bitonic sort: key descending, index ascending on ties
    for (int k = 2; k <= 1024; k <<= 1) {
        for (int j = k >> 1; j > 0; j >>= 1) {
            int ixj = tid ^ j;
            if (ixj > tid) {
                bool up = ((tid & k) == 0);  // "up" = best-first
                unsigned ka = skey[tid], kb = skey[ixj];
                int ia = sidx[tid], ib = sidx[ixj];
                bool bBetter = (kb > ka) || (kb == ka && ib < ia);
                if (up == bBetter) {
                    skey[tid] = kb; skey[ixj] = ka;
                    sidx[tid] = ib; sidx[ixj] = ia;
                }
            }
            __syncthreads();
        }
    }
    // gated output: out[r] = x3[idx[r]] * score[idx[r]]
    for (int t = tid; t < TOPK * 64; t += 1024) {
        int r = t >> 6, c = t & 63;
        int node = sidx[r];
        float v = 0.0f;
        if (node < n) v = x3[(size_t)node * 64 + c] * scores[node];
        out[t] = v;
    }
}

// ---------------------------------------------------------------------------
extern "C" void kernel_launch(void* const* d_in, const int* in_sizes, int n_in,
                              void* d_out, int out_size, void* d_ws, size_t ws_size,
                              hipStream_t stream) {
    const float* pos    = (const float*)d_in[0];
    const int*   ei     = (const int*)d_in[1];
    const float* c1_lw  = (const float*)d_in[2];
    const float* c1_lb  = (const float*)d_in[3];
    const float* c1_w1  = (const float*)d_in[4];
    const float* c1_b1  = (const float*)d_in[5];
    const float* c1_g1  = (const float*)d_in[6];
    const float* c1_be1 = (const float*)d_in[7];
    const float* c1_w2  = (const float*)d_in[8];
    const float* c1_b2  = (const float*)d_in[9];
    const float* c2_lw  = (const float*)d_in[10];
    const float* c2_lb  = (const float*)d_in[11];
    const float* c2_w1  = (const float*)d_in[12];
    const float* c2_b1  = (const float*)d_in[13];
    const float* c2_g1  = (const float*)d_in[14];
    const float* c2_be1 = (const float*)d_in[15];
    const float* c2_w2  = (const float*)d_in[16];
    const float* c2_b2  = (const float*)d_in[17];
    const float* pool_w = (const float*)d_in[18];

    float* ws = (float*)d_ws;
    const size_t R = (size_t)GN * 64;
    float* r0 = ws;               // x1 (N*16) then xl (N*64)
    float* r1 = ws + R;           // segmax; later h2 spans r1..r2 (N*128)
    float* r2 = ws + 2 * R;       // denom
    float* r3 = ws + 3 * R;       // num -> out2 (in place) -> x3
    float* r4 = ws + 4 * R;       // h1 (N*32)
    float* r5 = r4 + (size_t)GN * 32;  // x2 (N*16); later scores(N) + keys(N)
    float* st = r5 + (size_t)GN * 16;  // stats: sum[128], sq[128], bnA[128], bnB[128], misc
    float* bnSum = st, *bnSq = st + 128, *bnA = st + 256, *bnB = st + 384;
    float* invnorm = st + 512;
    float* scores = r5;
    unsigned* keys = (unsigned*)(r5 + GN);

    const int NB = (GN + 255) / 256;
    const int EB = (GE + 255) / 256;

    // ---- Layer 1 (C=16) ----
    hipMemsetAsync(r1, 0, (size_t)GN * 16 * 4, stream);   // segmax1 (0 < any msg)
    hipMemsetAsync(r2, 0, (size_t)GN * 16 * 4, stream);
    hipMemsetAsync(r3, 0, (size_t)GN * 16 * 4, stream);
    hipMemsetAsync(st, 0, 256 * 4, stream);

    k_lin1<<<NB, 256, 0, stream>>>(pos, c1_lw, c1_lb, r0, GN);
    k_edge_max<16><<<EB, 256, 0, stream>>>(ei, r0, (int*)r1, GE);
    k_edge_acc<16><<<EB, 256, 0, stream>>>(ei, r0, r1, r2, r3, GE);
    k_node1<<<NB, 256, 0, stream>>>(r0, r3, r2, c1_w1, c1_b1, r4, bnSum, bnSq, GN);
    k_bnfin<32><<<1, 32, 0, stream>>>(bnSum, bnSq, c1_g1, c1_be1, bnA, bnB, GN);
    k_node1b<<<NB, 256, 0, stream>>>(r4, bnA, bnB, c1_w2, c1_b2, r5, GN);

    // ---- Layer 2 (C=64) ----
    hipMemsetAsync(r1, 0, R * 4, stream);
    hipMemsetAsync(r2, 0, R * 4, stream);
    hipMemsetAsync(r3, 0, R * 4, stream);
    hipMemsetAsync(st, 0, 256 * 4, stream);

    // lin_src: xl[N,64] = x2[N,16] @ c2_lw + c2_lb   (WMMA f32)
    {
        int tiles = (GN / 16) * (64 / 16);
        k_wmma_gemm<false, false><<<(tiles + 7) / 8, 256, 0, stream>>>(
            r5, c2_lw, c2_lb, r0, 16, 64, nullptr, nullptr, tiles);
    }
    k_edge_max<64><<<EB, 256, 0, stream>>>(ei, r0, (int*)r1, GE);
    k_edge_acc<64><<<EB, 256, 0, stream>>>(ei, r0, r1, r2, r3, GE);
    k_aggr2<<<(int)((R + 255) / 256), 256, 0, stream>>>(r3, r2, r0, (int)R);

    // h2[N,128] = out2[N,64] @ c2_w1 + c2_b1   (WMMA f32)
    {
        int tiles = (GN / 16) * (128 / 16);
        k_wmma_gemm<false, false><<<(tiles + 7) / 8, 256, 0, stream>>>(
            r3, c2_w1, c2_b1, r1, 64, 128, nullptr, nullptr, tiles);
    }
    k_bnstats128<<<(GN + 127) / 128, 256, 0, stream>>>(r1, bnSum, bnSq, GN);
    k_bnfin<128><<<1, 128, 0, stream>>>(bnSum, bnSq, c2_g1, c2_be1, bnA, bnB, GN);

    // x3[N,64] = relu(bn(h2)) @ c2_w2 + c2_b2, then LeakyReLU   (WMMA f32, fused)
    {
        int tiles = (GN / 16) * (64 / 16);
        k_wmma_gemm<true, true><<<(tiles + 7) / 8, 256, 0, stream>>>(
            r1, c2_w2, c2_b2, r3, 128, 64, bnA, bnB, tiles);
    }

    // ---- TopK pooling ----
    k_wnorm<<<1, 64, 0, stream>>>(pool_w, invnorm);
    k_score<<<NB, 256, 0, stream>>>(r3, pool_w, invnorm, scores, keys, GN);
    k_topk<<<1, 1024, 0, stream>>>(keys, scores, r3, (float*)d_out, GN);
}